// MaSIF_ligand_57827439673345
// MI455X (gfx1250) — compile-verified
//
#include <hip/hip_runtime.h>
#include <hip/hip_bf16.h>

// ---------------------------------------------------------------------------
// MaSIF-ligand forward on gfx1250 (CDNA5, wave32).
//  K1: gaussian soft-binning + vertex-normalized descriptors (VALU/v_exp bound)
//  K2: conv einsum + max-over-16-rotations + bn1 + relu   (v_wmma_f32_16x16x4_f32)
//  K3: x @ W1 + b1 + bn2 + relu                           (wmma)
//  K4: covariance  cov[b] = X^T X / P                     (wmma)
//  K5: cov @ W2 + b2 + relu                               (wmma)
//  K6: bn3 + @W3 + b3  (tiny, scalar)
// ---------------------------------------------------------------------------

#define B_    32
#define P_    32
#define V_    200
#define NF_   4
#define TR_   80
#define NROT_ 16
#define BP_   (B_ * P_)

static constexpr float TWO_PI_F = 6.28318530717958647692f;
static constexpr float LOG2E_F  = 1.44269504088896340736f;

typedef __attribute__((ext_vector_type(2))) float v2f;
typedef __attribute__((ext_vector_type(8))) float v8f;

// ---------------------------------------------------------------------------
// Kernel 1: per-pocket gaussian binning.
// grid = BP_ blocks, block = 320 threads (thread = (f,t) bin), 10 wave32/WG.
// desc layout: [f][rot][bp][t]   (rows of the per-(f,rot) GEMM are contiguous)
// ---------------------------------------------------------------------------
__global__ __launch_bounds__(320) void k1_gauss_desc(
    const float* __restrict__ feat,    // [BP][V][NF]
    const float* __restrict__ rho,     // [BP][V]
    const float* __restrict__ theta,   // [BP][V]
    const float* __restrict__ mask,    // [BP][V]
    const float* __restrict__ mu_rho,  // [NF*TR]
    const float* __restrict__ sig_rho,
    const float* __restrict__ mu_th,
    const float* __restrict__ sig_th,
    float* __restrict__ desc)          // [NF][NROT][BP][TR]
{
    __shared__ float s_rho[V_], s_th[V_], s_msk[V_], s_feat[V_ * NF_];
    const int bp  = blockIdx.x;
    const int tid = threadIdx.x;                  // 0..319  == f*TR + t

    for (int i = tid; i < V_; i += 320) {
        s_rho[i] = rho[bp * V_ + i];
        s_th[i]  = theta[bp * V_ + i];
        s_msk[i] = mask[bp * V_ + i];
    }
    for (int i = tid; i < V_ * NF_; i += 320)
        s_feat[i] = feat[bp * V_ * NF_ + i];
    __syncthreads();

    const int f = tid / TR_;
    const int t = tid % TR_;

    const float mr  = mu_rho[tid];
    const float mt  = mu_th[tid];
    const float srv = sig_rho[tid];
    const float stv = sig_th[tid];
    // fold log2(e) into 1/(sigma^2+eps) so g = exp2(-(dr^2*isr + dt^2*ist))
    const float isr = LOG2E_F / (srv * srv + 1e-5f);
    const float ist = LOG2E_F / (stv * stv + 1e-5f);

    float S[NROT_], D[NROT_];
#pragma unroll
    for (int r = 0; r < NROT_; ++r) { S[r] = 0.f; D[r] = 0.f; }

    const float doff = TWO_PI_F / (float)NROT_;

    for (int v = 0; v < V_; ++v) {
        const float rv = s_rho[v];
        const float tv = s_th[v];
        const float mv = s_msk[v];
        const float fv = s_feat[v * NF_ + f];
        const float dr = rv - mr;
        const float ar = dr * dr * isr;           // rotation-invariant part
#pragma unroll
        for (int r = 0; r < NROT_; ++r) {
            float th = tv + doff * (float)r;      // in [0, 2*2pi)
            th = (th >= TWO_PI_F) ? th - TWO_PI_F : th;   // == mod(.., 2pi)
            const float dt = th - mt;
            const float a  = ar + dt * dt * ist;
            const float g  = __builtin_amdgcn_exp2f(-a) * mv;   // v_exp_f32
            S[r] += g;
            D[r] += g * fv;
        }
    }
#pragma unroll
    for (int r = 0; r < NROT_; ++r) {
        // g /= (sum_v g + eps)  then  sum_v g*feat  ==  D / (S + eps)
        desc[((f * NROT_ + r) * BP_ + bp) * TR_ + t] = D[r] / (S[r] + 1e-5f);
    }
}

// ---------------------------------------------------------------------------
// WMMA fragment helpers (V_WMMA_F32_16X16X4_F32, wave32):
//  A 16x4 : lanes 0-15 M=lane  K={k,k+1};  lanes16-31 M=lane-16 K={k+2,k+3}
//  B 4x16 : lanes 0-15 N=lane  K={k,k+1};  lanes16-31 N=lane-16 K={k+2,k+3}
//  C 16x16: VGPR r -> M = r (lanes 0-15) / r+8 (lanes 16-31), N = lane&15
// ---------------------------------------------------------------------------

// Kernel 2: conv = desc @ W_conv[f], max over rotations, +b_conv, bn1, relu.
// grid = (64 Mtiles, 5 Ntiles, 4 f); block = 32 (one wave, EXEC all ones).
__global__ __launch_bounds__(32) void k2_conv_max_bn1(
    const float* __restrict__ desc,    // [NF][NROT][BP][TR]
    const float* __restrict__ Wc,      // [NF][TR][TR]
    const float* __restrict__ bc,      // [NF][TR]
    const float* __restrict__ g1, const float* __restrict__ bb1,
    const float* __restrict__ m1, const float* __restrict__ vv1,
    float* __restrict__ x1)            // [BP][NF*TR]
{
    const int Mt = blockIdx.x, Nt = blockIdx.y, f = blockIdx.z;
    const int lane = threadIdx.x;
    const int m    = lane & 15;
    const int hi   = lane >> 4;
    const int koff = hi * 2;
    const int col  = Nt * 16 + m;
    const int row  = Mt * 16 + m;

    // Preload all 20 K-step B fragments of W_conv[f] (reused across 16 rots).
    v2f bfr[20];
#pragma unroll
    for (int ks = 0; ks < 20; ++ks) {
        const int k = ks * 4 + koff;
        bfr[ks].x = Wc[(f * TR_ + k    ) * TR_ + col];
        bfr[ks].y = Wc[(f * TR_ + k + 1) * TR_ + col];
    }

    float best[8];
#pragma unroll
    for (int i = 0; i < 8; ++i) best[i] = -3.4e38f;

    for (int r = 0; r < NROT_; ++r) {
        const float* dptr = desc + ((size_t)(f * NROT_ + r) * BP_ + row) * TR_;
        v8f c = {};
#pragma unroll
        for (int ks = 0; ks < 20; ++ks) {
            v2f a;
            a.x = dptr[ks * 4 + koff];
            a.y = dptr[ks * 4 + koff + 1];
            c = __builtin_amdgcn_wmma_f32_16x16x4_f32(
                    false, a, false, bfr[ks], (short)0, c, false, false);
        }
#pragma unroll
        for (int i = 0; i < 8; ++i) best[i] = fmaxf(best[i], c[i]);
    }

    const int   ch    = f * TR_ + col;            // channel after reshape
    const float scale = g1[ch] * rsqrtf(vv1[ch] + 1e-3f);
    const float bias  = bc[ch];                   // b_conv[f][col]
    const float mean  = m1[ch];
    const float beta  = bb1[ch];
#pragma unroll
    for (int i = 0; i < 8; ++i) {
        const int orow = Mt * 16 + i + 8 * hi;
        const float y = fmaxf((best[i] + bias - mean) * scale + beta, 0.f);
        x1[(size_t)orow * (NF_ * TR_) + ch] = y;
    }
}

// Kernel 3: x2 = relu(bn2(x1 @ W1 + b1)).  grid = (64, 5); block = 32.
__global__ __launch_bounds__(32) void k3_fc1_bn2(
    const float* __restrict__ x1,      // [BP][320]
    const float* __restrict__ W1,      // [320][80]
    const float* __restrict__ b1v,     // [80]
    const float* __restrict__ g2, const float* __restrict__ bb2,
    const float* __restrict__ m2, const float* __restrict__ vv2,
    float* __restrict__ x2)            // [BP][80]
{
    const int Mt = blockIdx.x, Nt = blockIdx.y;
    const int lane = threadIdx.x;
    const int m    = lane & 15;
    const int hi   = lane >> 4;
    const int koff = hi * 2;
    const int col  = Nt * 16 + m;
    const int row  = Mt * 16 + m;

    v8f c = {};
#pragma unroll 4
    for (int ks = 0; ks < 80; ++ks) {             // K = 320
        const int k = ks * 4 + koff;
        v2f a, b;
        a.x = x1[(size_t)row * 320 + k];
        a.y = x1[(size_t)row * 320 + k + 1];
        b.x = W1[(size_t)k * TR_ + col];
        b.y = W1[(size_t)(k + 1) * TR_ + col];
        c = __builtin_amdgcn_wmma_f32_16x16x4_f32(
                false, a, false, b, (short)0, c, false, false);
    }

    const float scale = g2[col] * rsqrtf(vv2[col] + 1e-3f);
    const float bias  = b1v[col];
    const float mean  = m2[col];
    const float beta  = bb2[col];
#pragma unroll
    for (int i = 0; i < 8; ++i) {
        const int orow = Mt * 16 + i + 8 * hi;
        const float y = fmaxf((c[i] + bias - mean) * scale + beta, 0.f);
        x2[(size_t)orow * TR_ + col] = y;
    }
}

// Kernel 4: cov[b] = X^T X / P,  X = x2[b*P .. b*P+31][80].
// grid = (5 Itiles, 5 Jtiles, 32 batches); block = 32.  K = P = 32.
__global__ __launch_bounds__(32) void k4_cov(
    const float* __restrict__ x2,      // [BP][80]
    float* __restrict__ cov)           // [B][80][80]
{
    const int It = blockIdx.x, Jt = blockIdx.y, b = blockIdx.z;
    const int lane = threadIdx.x;
    const int m    = lane & 15;
    const int hi   = lane >> 4;
    const int koff = hi * 2;
    const int i    = It * 16 + m;
    const int j    = Jt * 16 + m;

    v8f c = {};
#pragma unroll
    for (int ks = 0; ks < 8; ++ks) {              // K = 32 vertices-in-pocket
        const int p = ks * 4 + koff;
        v2f a, bb;
        a.x  = x2[(size_t)(b * P_ + p    ) * TR_ + i];   // A[i][p] = X^T
        a.y  = x2[(size_t)(b * P_ + p + 1) * TR_ + i];
        bb.x = x2[(size_t)(b * P_ + p    ) * TR_ + j];   // B[p][j] = X
        bb.y = x2[(size_t)(b * P_ + p + 1) * TR_ + j];
        c = __builtin_amdgcn_wmma_f32_16x16x4_f32(
                false, a, false, bb, (short)0, c, false, false);
    }
    const float inv_p = 1.0f / (float)P_;
#pragma unroll
    for (int r = 0; r < 8; ++r) {
        const int oi = It * 16 + r + 8 * hi;
        cov[(size_t)b * (TR_ * TR_) + (size_t)oi * TR_ + (Jt * 16 + m)] =
            c[r] * inv_p;
    }
}

// Kernel 5: y = relu(cov_flat @ W2 + b2).  grid = (2 Mtiles, 4 Ntiles).
__global__ __launch_bounds__(32) void k5_fc2(
    const float* __restrict__ cov,     // [B][6400]
    const float* __restrict__ W2,      // [6400][64]
    const float* __restrict__ b2v,     // [64]
    float* __restrict__ y)             // [B][64]
{
    const int Mt = blockIdx.x, Nt = blockIdx.y;
    const int lane = threadIdx.x;
    const int m    = lane & 15;
    const int hi   = lane >> 4;
    const int koff = hi * 2;
    const int col  = Nt * 16 + m;
    const int row  = Mt * 16 + m;

    v8f c = {};
#pragma unroll 8
    for (int ks = 0; ks < 1600; ++ks) {           // K = 6400
        const int k = ks * 4 + koff;
        v2f a, b;
        a.x = cov[(size_t)row * 6400 + k];
        a.y = cov[(size_t)row * 6400 + k + 1];
        b.x = W2[(size_t)k * 64 + col];
        b.y = W2[(size_t)(k + 1) * 64 + col];
        c = __builtin_amdgcn_wmma_f32_16x16x4_f32(
                false, a, false, b, (short)0, c, false, false);
    }
    const float bias = b2v[col];
#pragma unroll
    for (int i = 0; i < 8; ++i) {
        const int orow = Mt * 16 + i + 8 * hi;
        y[(size_t)orow * 64 + col] = fmaxf(c[i] + bias, 0.f);
    }
}

// Kernel 6: out = bn3(y) @ W3 + b3.  One block, 224 threads (b,l pairs).
__global__ __launch_bounds__(256) void k6_bn3_fc3(
    const float* __restrict__ y,       // [B][64]
    const float* __restrict__ g3, const float* __restrict__ bb3,
    const float* __restrict__ m3, const float* __restrict__ vv3,
    const float* __restrict__ W3,      // [64][7]
    const float* __restrict__ b3v,     // [7]
    float* __restrict__ out)           // [B][7]
{
    const int tid = threadIdx.x;
    if (tid >= B_ * 7) return;
    const int b = tid / 7;
    const int l = tid % 7;
    float acc = b3v[l];
#pragma unroll
    for (int k = 0; k < 64; ++k) {
        const float t = (y[b * 64 + k] - m3[k]) * (g3[k] * rsqrtf(vv3[k] + 1e-3f)) + bb3[k];
        acc += t * W3[k * 7 + l];
    }
    out[b * 7 + l] = acc;
}

// ---------------------------------------------------------------------------
extern "C" void kernel_launch(void* const* d_in, const int* in_sizes, int n_in,
                              void* d_out, int out_size, void* d_ws, size_t ws_size,
                              hipStream_t stream) {
    const float* feat    = (const float*)d_in[0];
    const float* rho     = (const float*)d_in[1];
    const float* theta   = (const float*)d_in[2];
    const float* mask    = (const float*)d_in[3];
    const float* mu_rho  = (const float*)d_in[4];
    const float* sig_rho = (const float*)d_in[5];
    const float* mu_th   = (const float*)d_in[6];
    const float* sig_th  = (const float*)d_in[7];
    const float* Wc      = (const float*)d_in[8];
    const float* bc      = (const float*)d_in[9];
    const float* W1      = (const float*)d_in[10];
    const float* b1v     = (const float*)d_in[11];
    const float* g1      = (const float*)d_in[12];
    const float* bb1     = (const float*)d_in[13];
    const float* m1      = (const float*)d_in[14];
    const float* vv1     = (const float*)d_in[15];
    const float* g2      = (const float*)d_in[16];
    const float* bb2     = (const float*)d_in[17];
    const float* m2      = (const float*)d_in[18];
    const float* vv2     = (const float*)d_in[19];
    const float* W2      = (const float*)d_in[20];
    const float* b2v     = (const float*)d_in[21];
    const float* g3      = (const float*)d_in[22];
    const float* bb3     = (const float*)d_in[23];
    const float* m3      = (const float*)d_in[24];
    const float* vv3     = (const float*)d_in[25];
    const float* W3      = (const float*)d_in[26];
    const float* b3v     = (const float*)d_in[27];

    // Workspace partition (floats).  Total ~23.5 MB.
    float* ws    = (float*)d_ws;
    float* desc  = ws;                                   // [4][16][1024][80] = 5,242,880
    float* x1    = desc + (size_t)NF_ * NROT_ * BP_ * TR_;   // [1024][320] = 327,680
    float* x2    = x1   + (size_t)BP_ * NF_ * TR_;           // [1024][80]  =  81,920
    float* cov   = x2   + (size_t)BP_ * TR_;                 // [32][6400]  = 204,800
    float* ybuf  = cov  + (size_t)B_ * TR_ * TR_;            // [32][64]    =   2,048
    float* outf  = (float*)d_out;

    k1_gauss_desc<<<dim3(BP_), dim3(320), 0, stream>>>(
        feat, rho, theta, mask, mu_rho, sig_rho, mu_th, sig_th, desc);

    k2_conv_max_bn1<<<dim3(BP_ / 16, TR_ / 16, NF_), dim3(32), 0, stream>>>(
        desc, Wc, bc, g1, bb1, m1, vv1, x1);

    k3_fc1_bn2<<<dim3(BP_ / 16, TR_ / 16), dim3(32), 0, stream>>>(
        x1, W1, b1v, g2, bb2, m2, vv2, x2);

    k4_cov<<<dim3(TR_ / 16, TR_ / 16, B_), dim3(32), 0, stream>>>(x2, cov);

    k5_fc2<<<dim3(B_ / 16, 64 / 16), dim3(32), 0, stream>>>(cov, W2, b2v, ybuf);

    k6_bn3_fc3<<<dim3(1), dim3(256), 0, stream>>>(
        ybuf, g3, bb3, m3, vv3, W3, b3v, outf);
}